// MaskedAdaptiveL1ChamferLoss_12446815224305
// MI455X (gfx1250) — compile-verified
//
#include <hip/hip_runtime.h>
#include <math.h>

// ---------------------------------------------------------------------------
// MaskedAdaptiveL1ChamferLoss for MI455X (gfx1250, wave32)
//
// Roofline: inputs total ~640KB (sits in 192MB L2); dominant work is
// 2 * 8 * 2048^2 L1 pairwise distances (~470M VALU lane-ops) -> VALU-bound.
// L1 distance is not bilinear, so the pairwise core cannot be a WMMA GEMM.
// CDNA5-specific paths used:
//   * TDM tensor_load_to_lds + s_wait_tensorcnt to stage the mask row in LDS
//   * ds_load_b128 broadcasts of the staged float4 database points
//   * V_WMMA_F32_16X16X4_F32 (A = ones) as an exact-f32 block-sum engine:
//     D[i][n] = sum_k B[k][n] + C[i][n]; chaining 4 WMMAs sums 256 values.
//     Summing all 256 D slots = 16x the total (rows identical), so *1/16
//     (exact pow2 scale) recovers the block sum independent of VGPR layout.
// ---------------------------------------------------------------------------

#define B_DIM  8
#define N_DIM  2048
#define TPB    256
#define TILES  (N_DIM / TPB)   // 8 row-tiles per batch
#define BIG_F  1e30f

typedef unsigned int u32x4 __attribute__((ext_vector_type(4)));
typedef int          i32x8 __attribute__((ext_vector_type(8)));
typedef int          i32x4 __attribute__((ext_vector_type(4)));
typedef float        v2f   __attribute__((ext_vector_type(2)));
typedef float        v8f   __attribute__((ext_vector_type(8)));

#if defined(__AMDGCN__) && __has_builtin(__builtin_amdgcn_tensor_load_to_lds) && \
    __has_builtin(__builtin_amdgcn_s_wait_tensorcnt)
#define HAVE_TDM 1
#else
#define HAVE_TDM 0
#endif

#if defined(__AMDGCN__) && __has_builtin(__builtin_amdgcn_wmma_f32_16x16x4_f32)
#define HAVE_WMMA_F32 1
#else
#define HAVE_WMMA_F32 0
#endif

// ---------------------------------------------------------------------------
// Deterministic block-wide sum of 256 per-thread values using f32 WMMA.
// All threads of the block must call this; tid0 stores the result to *gout.
// red must be a 256-float LDS buffer.
// ---------------------------------------------------------------------------
__device__ __forceinline__ void block_sum_store(float val, float* red, int tid,
                                                float* gout)
{
    __syncthreads();           // make red reusable across calls
    red[tid] = val;
    __syncthreads();
#if HAVE_WMMA_F32
    if (tid < 32) {            // wave 0, EXEC all ones (WMMA requirement)
        v8f acc = {0.f, 0.f, 0.f, 0.f, 0.f, 0.f, 0.f, 0.f};
        v2f ones = {1.0f, 1.0f};
#pragma unroll
        for (int k = 0; k < 4; ++k) {
            v2f bb;
            bb.x = red[(2 * k + 0) * 32 + tid];
            bb.y = red[(2 * k + 1) * 32 + tid];
            // D = ones(16x4) * B(4x16) + C  -> column sums accumulate in C
            acc = __builtin_amdgcn_wmma_f32_16x16x4_f32(
                false, ones, false, bb, (short)0, acc, false, false);
        }
        // Every D row is identical -> sum of all 256 slots == 16 * total.
        float s = acc[0] + acc[1] + acc[2] + acc[3] +
                  acc[4] + acc[5] + acc[6] + acc[7];
        red[tid] = s;
    }
    __syncthreads();
    if (tid == 0) {
        float t = 0.0f;
        for (int l = 0; l < 32; ++l) t += red[l];
        *gout = t * 0.0625f;   // exact /16
    }
#else
    for (int stride = TPB / 2; stride > 0; stride >>= 1) {
        if (tid < stride) red[tid] += red[tid + stride];
        __syncthreads();
    }
    if (tid == 0) *gout = red[0];
#endif
}

// ---------------------------------------------------------------------------
// Kernel 1: masked L1 term + per-batch mask counts.
// l1part[b] = sum_n mean_c(|pred-target|) * mask ;  cntpart[b] = sum_n mask
// ---------------------------------------------------------------------------
__global__ __launch_bounds__(TPB)
void l1_kernel(const float* __restrict__ pred,
               const float* __restrict__ target,
               const int*  __restrict__ mask,
               float* __restrict__ l1part,
               float* __restrict__ cntpart)
{
    __shared__ float red[TPB];

    const int b   = blockIdx.x;
    const int tid = threadIdx.x;

    float s = 0.0f, c = 0.0f;
    for (int n = tid; n < N_DIM; n += TPB) {
        const size_t i2 = (size_t)b * N_DIM + n;
        const size_t i3 = i2 * 3;
        const float m = (float)mask[i2];
        const float d = (fabsf(pred[i3 + 0] - target[i3 + 0]) +
                         fabsf(pred[i3 + 1] - target[i3 + 1]) +
                         fabsf(pred[i3 + 2] - target[i3 + 2])) * (1.0f / 3.0f);
        s += d * m;
        c += m;
    }
    block_sum_store(s, red, tid, &l1part[b]);
    block_sum_store(c, red, tid, &cntpart[b]);
}

// ---------------------------------------------------------------------------
// Kernel 2: one-directional masked nearest-neighbor (L1) partial sums.
//   dir==0: query = clean = points+target, database = predp = points+pred
//   dir==1: query = predp, database = clean
// db point carries bias = mask ? 0 : 1e30; distances O(10) << ulp(1e30), so
// d + bias == 1e30 exactly for invalid points == jnp.where(mask, D, BIG).
// ---------------------------------------------------------------------------
__global__ __launch_bounds__(TPB)
void chamfer_dir_kernel(const float* __restrict__ pred,
                        const float* __restrict__ target,
                        const int*  __restrict__ mask,
                        const float* __restrict__ points,
                        float* __restrict__ dirpart)
{
    __shared__ float4 db[N_DIM];    // 32 KB: database points + mask bias
    __shared__ int    smask[N_DIM]; //  8 KB: mask row for this batch
    __shared__ float  red[TPB];

    const int tile = blockIdx.x;   // 0..7  (row tile)
    const int b    = blockIdx.y;   // 0..7  (batch)
    const int dir  = blockIdx.z;   // 0..1  (chamfer direction)
    const int tid  = threadIdx.x;

    const float* srcQ  = dir ? pred   : target;  // query  = points + srcQ
    const float* srcDB = dir ? target : pred;    // db     = points + srcDB

    const size_t base2 = (size_t)b * N_DIM;
    const size_t base3 = base2 * 3;

    // Warm the L2->WGP path for the point data we are about to stream.
    __builtin_prefetch(points + base3, 0, 0);
    __builtin_prefetch(srcDB + base3, 0, 0);

    // ---- Stage mask row for this batch into LDS (TDM async DMA) ---------
#if HAVE_TDM
    if (tid < 32) {  // wave 0 issues the TDM descriptor (per-wave op)
        const unsigned lds_addr =
            (unsigned)(unsigned long long)(const void*)&smask[0];
        const unsigned long long gaddr =
            (unsigned long long)(const void*)(mask + base2);
        u32x4 g0;
        g0.x = 1u;                                        // count=1, user desc
        g0.y = lds_addr;                                  // LDS byte address
        g0.z = (unsigned)(gaddr & 0xFFFFFFFFull);         // global addr lo
        g0.w = (unsigned)((gaddr >> 32) & 0x01FFFFFFull)  // global addr hi
             | (2u << 30);                                // type = 2 (image)
        i32x8 g1;
        g1[0] = (int)(2u << 16);        // data_size = 4B, no multicast
        g1[1] = (int)(N_DIM << 16);     // tensor_dim0 low16 = 2048
        g1[2] = (int)(1u << 16);        // tensor_dim1 = 1
        g1[3] = (int)(N_DIM << 16);     // tile_dim0 = 2048
        g1[4] = 1;                      // tile_dim1 = 1, tile_dim2 = 0
        g1[5] = N_DIM;                  // tensor_dim0_stride = 2048
        g1[6] = 0;
        g1[7] = 0;
        i32x4 g2 = {0, 0, 0, 0};
        i32x4 g3 = {0, 0, 0, 0};
#if __clang_major__ >= 23
        i32x8 g4 = {0, 0, 0, 0, 0, 0, 0, 0};
        __builtin_amdgcn_tensor_load_to_lds(g0, g1, g2, g3, g4, 0);
#else
        __builtin_amdgcn_tensor_load_to_lds(g0, g1, g2, g3, 0);
#endif
        __builtin_amdgcn_s_wait_tensorcnt(0);
    }
#else
    for (int j = tid; j < N_DIM; j += TPB) smask[j] = mask[base2 + j];
#endif
    __syncthreads();

    // ---- Stage database points (x, y, z, bias) into LDS -----------------
    for (int j = tid; j < N_DIM; j += TPB) {
        const size_t i3 = base3 + 3 * (size_t)j;
        const float px = points[i3 + 0] + srcDB[i3 + 0];
        const float py = points[i3 + 1] + srcDB[i3 + 1];
        const float pz = points[i3 + 2] + srcDB[i3 + 2];
        const float bias = smask[j] ? 0.0f : BIG_F;
        db[j] = make_float4(px, py, pz, bias);
    }
    __syncthreads();

    // ---- Each thread: masked min over all 2048 database points ----------
    const int i  = tile * TPB + tid;
    const size_t qi3 = base3 + 3 * (size_t)i;
    const float qx = points[qi3 + 0] + srcQ[qi3 + 0];
    const float qy = points[qi3 + 1] + srcQ[qi3 + 1];
    const float qz = points[qi3 + 2] + srcQ[qi3 + 2];
    const float qw = (float)smask[i];  // query-side mask weight

    float m0 = BIG_F, m1 = BIG_F, m2 = BIG_F, m3 = BIG_F;
    for (int j = 0; j < N_DIM; j += 4) {
        const float4 p0 = db[j + 0];
        const float4 p1 = db[j + 1];
        const float4 p2 = db[j + 2];
        const float4 p3 = db[j + 3];
        m0 = fminf(m0, fabsf(qx - p0.x) + fabsf(qy - p0.y) + fabsf(qz - p0.z) + p0.w);
        m1 = fminf(m1, fabsf(qx - p1.x) + fabsf(qy - p1.y) + fabsf(qz - p1.z) + p1.w);
        m2 = fminf(m2, fabsf(qx - p2.x) + fabsf(qy - p2.y) + fabsf(qz - p2.z) + p2.w);
        m3 = fminf(m3, fabsf(qx - p3.x) + fabsf(qy - p3.y) + fabsf(qz - p3.z) + p3.w);
    }
    const float mn = fminf(fminf(m0, m1), fminf(m2, m3));

    // ---- WMMA-based deterministic block reduction of mn * qw -------------
    block_sum_store(mn * qw, red, tid,
                    &dirpart[((dir * B_DIM) + b) * TILES + tile]);
}

// ---------------------------------------------------------------------------
// Kernel 3: final deterministic combine (tiny, single thread).
// ws layout (floats): [0..127] dirpart, [128..135] l1part, [136..143] cntpart
// ---------------------------------------------------------------------------
__global__ void final_kernel(const float* __restrict__ ws, float* __restrict__ out)
{
    float l1sum = 0.0f, msum = 0.0f;
    for (int b = 0; b < B_DIM; ++b) {
        l1sum += ws[128 + b];
        msum  += ws[136 + b];
    }
    const float l1 = l1sum / msum;

    float cd = 0.0f;
    for (int b = 0; b < B_DIM; ++b) {
        float sx = 0.0f, sy = 0.0f;
        for (int t = 0; t < TILES; ++t) {
            sx += ws[(0 * B_DIM + b) * TILES + t];  // (min_x * mask).sum
            sy += ws[(1 * B_DIM + b) * TILES + t];  // (min_y * mask).sum
        }
        const float cnt = ws[136 + b];
        cd += (sx + sy) / cnt;
    }
    cd *= (1.0f / (float)B_DIM);

    out[0] = l1 + expf(-l1) * cd;
}

// ---------------------------------------------------------------------------
extern "C" void kernel_launch(void* const* d_in, const int* in_sizes, int n_in,
                              void* d_out, int out_size, void* d_ws, size_t ws_size,
                              hipStream_t stream)
{
    (void)in_sizes; (void)n_in; (void)out_size; (void)ws_size;

    const float* pred   = (const float*)d_in[0];
    const float* target = (const float*)d_in[1];
    const int*   mask   = (const int*)d_in[2];
    const float* points = (const float*)d_in[3];

    float* wsf     = (float*)d_ws;
    float* dirpart = wsf;         // 128 floats
    float* l1part  = wsf + 128;   //   8 floats
    float* cntpart = wsf + 136;   //   8 floats

    hipLaunchKernelGGL(l1_kernel, dim3(B_DIM), dim3(TPB), 0, stream,
                       pred, target, mask, l1part, cntpart);
    hipLaunchKernelGGL(chamfer_dir_kernel, dim3(TILES, B_DIM, 2), dim3(TPB), 0, stream,
                       pred, target, mask, points, dirpart);
    hipLaunchKernelGGL(final_kernel, dim3(1), dim3(1), 0, stream,
                       wsf, (float*)d_out);
}